// RobotCollision_65584150610357
// MI455X (gfx1250) — compile-verified
//
#include <hip/hip_runtime.h>

// Problem constants (match reference: pos [T=64, N=1024, 3] f32, rad=0.15)
#define T_STEPS 64
#define NPTS    1024
#define TILE    16
#define NTILES  (NPTS / TILE)                  // 64
#define NPAIRS  (NTILES * (NTILES + 1) / 2)    // 2080 upper-tri tile pairs
#define WAVES_PER_BLOCK 8
#define BLOCK   (WAVES_PER_BLOCK * 32)

typedef __attribute__((ext_vector_type(2))) float v2f;
typedef __attribute__((ext_vector_type(8))) float v8f;

__global__ void RobotCollision_zero_kernel(float* out) {
    if (threadIdx.x == 0) {
        out[0] = 0.0f;   // count (accumulated as exact-integer float)
        out[1] = 0.0f;   // loss
    }
}

__device__ __forceinline__ int tri_first(int r) {
    // # of tile pairs in rows < r  (row r spans tj = r..NTILES-1)
    return r * NTILES - (r * (r - 1)) / 2;
}

__launch_bounds__(BLOCK)
__global__ void RobotCollision_wmma_kernel(const float* __restrict__ pos,
                                           float* __restrict__ out) {
    const float MIND = 0.30f;          // 2 * rad
    const float MIN2 = MIND * MIND;    // 0.09

    // ---- decode upper-triangular tile pair (ti <= tj) from blockIdx.x ----
    const int p = blockIdx.x;
    int ti = (int)((129.0f - __builtin_amdgcn_sqrtf((float)(129 * 129 - 8 * p))) * 0.5f);
    if (ti < 0) ti = 0;
    while (tri_first(ti + 1) <= p) ++ti;   // robust fixup vs f32 rounding
    while (tri_first(ti) > p) --ti;
    const int tj = ti + (p - tri_first(ti));

    const int lane = threadIdx.x & 31;
    const int wave = threadIdx.x >> 5;
    const int half = lane >> 4;        // 0: K={0,1}   1: K={2,3}
    const int l16  = lane & 15;

    const int gi = ti * TILE + l16;    // A-matrix row index for this lane
    const int gj = tj * TILE + l16;    // B-matrix col index for this lane

    float lloss = 0.0f;
    float lcnt  = 0.0f;

    // each wave handles timesteps t = wave, wave+8, ... (uniform loop -> EXEC all 1s)
    for (int t = wave; t < T_STEPS; t += WAVES_PER_BLOCK) {
        const float* pi = pos + ((size_t)t * NPTS + gi) * 3;
        const float* pj = pos + ((size_t)t * NPTS + gj) * 3;
        const float pix = pi[0], piy = pi[1], piz = pi[2];
        const float pjx = pj[0], pjy = pj[1], pjz = pj[2];

        const float ni_lane = pix * pix + piy * piy + piz * piz; // |P_i[l16]|^2
        const float nj_lane = pjx * pjx + pjy * pjy + pjz * pjz; // |P_j[l16]|^2

        // Fold the row norm into the spare K slot:
        //   A[m] = (x, y, z, |p_i|^2),  B[:,n] = (-2x, -2y, -2z, 1)
        //   => G[m][n] = |p_i[m]|^2 - 2 p_i[m].p_j[n],  dist^2 = G + |p_j[n]|^2
        // A 16x4 f32: lanes 0-15 -> {K0,K1}; lanes 16-31 -> {K2,K3}
        v2f a, b;
        a.x = half ? piz : pix;
        a.y = half ? ni_lane : piy;
        // B 4x16 f32: same K split, N = lane&15
        b.x = half ? (-2.0f * pjz) : (-2.0f * pjx);
        b.y = half ? 1.0f          : (-2.0f * pjy);

        v8f c = {};
        c = __builtin_amdgcn_wmma_f32_16x16x4_f32(
                /*neg_a=*/false, a, /*neg_b=*/false, b,
                /*c_mod=*/(short)0, c, /*reuse_a=*/false, /*reuse_b=*/false);

        // C/D layout: element r of v8f in lane L -> M = r + 8*(L>>4), N = L&15
#pragma unroll
        for (int r = 0; r < 8; ++r) {
            const int m  = r + 8 * half;          // row within tile i
            const int mg = ti * TILE + m;         // global i index
            const int ng = tj * TILE + l16;       // global j index
            const float d2 = c[r] + nj_lane;
            const bool hit = (mg < ng) & (d2 < MIN2);
            // safe sqrt (as in reference); raw v_sqrt_f32, no IEEE fixup sequence
            const float d   = __builtin_amdgcn_sqrtf(d2 > 0.0f ? d2 : 1.0f);
            const float pen = MIND - d;
            lloss += hit ? pen * pen : 0.0f;
            lcnt  += hit ? 1.0f : 0.0f;
        }
    }

    // ---- wave32 reduction ----
#pragma unroll
    for (int o = 16; o > 0; o >>= 1) {
        lloss += __shfl_xor(lloss, o, 32);
        lcnt  += __shfl_xor(lcnt,  o, 32);
    }

    // ---- block reduction (8 waves) -> 2 global atomics per block ----
    __shared__ float sL[WAVES_PER_BLOCK];
    __shared__ float sC[WAVES_PER_BLOCK];
    if (lane == 0) { sL[wave] = lloss; sC[wave] = lcnt; }
    __syncthreads();
    if (threadIdx.x == 0) {
        float L = 0.0f, C = 0.0f;
#pragma unroll
        for (int w = 0; w < WAVES_PER_BLOCK; ++w) { L += sL[w]; C += sC[w]; }
        atomicAdd(out + 0, C);
        atomicAdd(out + 1, L);
    }
}

extern "C" void kernel_launch(void* const* d_in, const int* in_sizes, int n_in,
                              void* d_out, int out_size, void* d_ws, size_t ws_size,
                              hipStream_t stream) {
    const float* pos = (const float*)d_in[0];  // [64, 1024, 3] f32
    float* out = (float*)d_out;                // [count, loss]
    (void)in_sizes; (void)n_in; (void)out_size; (void)d_ws; (void)ws_size;

    RobotCollision_zero_kernel<<<1, 32, 0, stream>>>(out);
    RobotCollision_wmma_kernel<<<NPAIRS, BLOCK, 0, stream>>>(pos, out);
}